// SparseGating_73289321939550
// MI455X (gfx1250) — compile-verified
//
#include <hip/hip_runtime.h>

// SparseGating forward: y = x (straight-through), mask = per-row top-k(|x|), k=614, D=4096.
// One 256-thread (8-wave) workgroup per row; radix-256 select over |x| bit patterns.

typedef float v4f __attribute__((ext_vector_type(4)));

#define DDIM   4096
#define KSEL   614          // round(0.15 * 4096)
#define NTHR   256
#define EPT    16           // elements per thread (DDIM / NTHR)
#define NWAVE  8            // wave32: 256 / 32
#define HPAD   264          // 256 bins + 8 pad words -> per-wave bank skew

__global__ __launch_bounds__(NTHR) void sparse_gate_topk(
    const float* __restrict__ x, float* __restrict__ y, float* __restrict__ mask)
{
    __shared__ unsigned hist[NWAVE][HPAD];
    __shared__ unsigned wsum[NWAVE];
    __shared__ unsigned eqw[NWAVE];
    __shared__ unsigned s_pick;     // (selected byte << 16) | remaining

    const int tid  = threadIdx.x;
    const int lane = tid & 31;
    const int wv   = tid >> 5;

    const size_t base = (size_t)blockIdx.x * (size_t)DDIM + (size_t)tid * EPT;

    // ---- load 16 contiguous floats per thread (4 x b128, non-temporal) ----
    const v4f* xp = reinterpret_cast<const v4f*>(x + base);
    v4f v0 = __builtin_nontemporal_load(xp + 0);
    v4f v1 = __builtin_nontemporal_load(xp + 1);
    v4f v2 = __builtin_nontemporal_load(xp + 2);
    v4f v3 = __builtin_nontemporal_load(xp + 3);

    unsigned bits[EPT];
#pragma unroll
    for (int i = 0; i < 4; ++i) {
        bits[i]      = __float_as_uint(v0[i]) & 0x7FFFFFFFu;
        bits[4 + i]  = __float_as_uint(v1[i]) & 0x7FFFFFFFu;
        bits[8 + i]  = __float_as_uint(v2[i]) & 0x7FFFFFFFu;
        bits[12 + i] = __float_as_uint(v3[i]) & 0x7FFFFFFFu;
    }

    // ---- y = x: fire the copy stores NOW, overlap with the select phase ----
    // (also kills v0..v3 early -> 16 fewer live VGPRs through the select)
    v4f* yp = reinterpret_cast<v4f*>(y + base);
    __builtin_nontemporal_store(v0, yp + 0);
    __builtin_nontemporal_store(v1, yp + 1);
    __builtin_nontemporal_store(v2, yp + 2);
    __builtin_nontemporal_store(v3, yp + 3);

    // ---- initial histogram clear ----
    {
        unsigned* hflat = &hist[0][0];
#pragma unroll
        for (int i = 0; i < (NWAVE * HPAD + NTHR - 1) / NTHR; ++i) {
            int idx = tid + i * NTHR;
            if (idx < NWAVE * HPAD) hflat[idx] = 0;
        }
    }
    __syncthreads();

    unsigned aliveMask = 0xFFFFu;   // per-element "still a threshold candidate"
    unsigned prefix    = 0;         // selected high bits of the threshold so far
    unsigned rem       = KSEL;      // how many still to take inside current prefix

    // ---- 4-pass radix-256 select (MSB -> LSB) over |x| bit patterns ----
#pragma unroll
    for (int pass = 0; pass < 4; ++pass) {
        const int shift = 24 - pass * 8;

        // per-wave histogram of the current byte (only surviving candidates)
#pragma unroll
        for (int i = 0; i < EPT; ++i) {
            if (aliveMask & (1u << i))
                atomicAdd(&hist[wv][(bits[i] >> shift) & 0xFFu], 1u);
        }
        __syncthreads();                              // bar A: hist complete

        // thread tid owns bin b = 255 - tid; ascending tid == descending bin.
        const unsigned b = 255u - (unsigned)tid;
        unsigned val = 0;
#pragma unroll
        for (int ww = 0; ww < NWAVE; ++ww) val += hist[ww][b];

        // wave32 inclusive scan (counts of bins >= b within this wave's chunk)
        unsigned inc = val;
#pragma unroll
        for (int off = 1; off < 32; off <<= 1) {
            unsigned n = __shfl_up(inc, off, 32);
            if (lane >= off) inc += n;
        }
        if (lane == 31) wsum[wv] = inc;
        __syncthreads();                              // bar B: all hist reads done

        unsigned woff = 0;
        for (int ww = 0; ww < wv; ++ww) woff += wsum[ww];
        const unsigned cnt_ge = woff + inc;   // candidates with byte >= b
        const unsigned cnt_gt = cnt_ge - val; // candidates with byte >  b
        if (cnt_gt < rem && rem <= cnt_ge)    // exactly one thread matches
            s_pick = (b << 16) | (rem - cnt_gt);

        // overlap: clear the histograms for the next pass (safe after bar B)
        if (pass < 3) {
            unsigned* hflat = &hist[0][0];
#pragma unroll
            for (int i = 0; i < (NWAVE * HPAD + NTHR - 1) / NTHR; ++i) {
                int idx = tid + i * NTHR;
                if (idx < NWAVE * HPAD) hflat[idx] = 0;
            }
        }
        __syncthreads();                              // bar C: pick + clear visible

        const unsigned pick    = s_pick;
        const unsigned selByte = pick >> 16;
        rem    = pick & 0xFFFFu;
        prefix |= selByte << shift;

#pragma unroll
        for (int i = 0; i < EPT; ++i) {
            if ((aliveMask & (1u << i)) && (((bits[i] >> shift) & 0xFFu) != selByte))
                aliveMask &= ~(1u << i);
        }
    }

    const unsigned Tb     = prefix;   // exact threshold bit pattern
    const unsigned keepEq = rem;      // # of == threshold elements to keep (>=1)

    // ---- stable rank among equals (index order == tid order by layout) ----
    unsigned eqc = 0;
#pragma unroll
    for (int i = 0; i < EPT; ++i) eqc += (bits[i] == Tb) ? 1u : 0u;

    unsigned einc = eqc;
#pragma unroll
    for (int off = 1; off < 32; off <<= 1) {
        unsigned n = __shfl_up(einc, off, 32);
        if (lane >= off) einc += n;
    }
    if (lane == 31) eqw[wv] = einc;
    __syncthreads();
    unsigned eoff = 0;
    for (int ww = 0; ww < wv; ++ww) eoff += eqw[ww];
    unsigned eqSeen = eoff + einc - eqc;  // equals strictly before this thread

    float m[EPT];
#pragma unroll
    for (int i = 0; i < EPT; ++i) {
        float mi = 0.0f;
        if (bits[i] > Tb) {
            mi = 1.0f;
        } else if (bits[i] == Tb) {
            if (eqSeen < keepEq) mi = 1.0f;   // lowest-index-first, like lax.top_k
            eqSeen++;
        }
        m[i] = mi;
    }

    // ---- mask stores ----
    v4f m0 = { m[0],  m[1],  m[2],  m[3]  };
    v4f m1 = { m[4],  m[5],  m[6],  m[7]  };
    v4f m2 = { m[8],  m[9],  m[10], m[11] };
    v4f m3 = { m[12], m[13], m[14], m[15] };
    v4f* mp = reinterpret_cast<v4f*>(mask + base);
    __builtin_nontemporal_store(m0, mp + 0);
    __builtin_nontemporal_store(m1, mp + 1);
    __builtin_nontemporal_store(m2, mp + 2);
    __builtin_nontemporal_store(m3, mp + 3);
}

extern "C" void kernel_launch(void* const* d_in, const int* in_sizes, int n_in,
                              void* d_out, int out_size, void* d_ws, size_t ws_size,
                              hipStream_t stream) {
    const float* x = (const float*)d_in[0];
    float*       y = (float*)d_out;            // output tuple: [y | mask], both f32
    const size_t n = (size_t)in_sizes[0];      // B*T*D = 67,108,864
    float*    mask = y + n;
    const int rows = (int)(n / DDIM);          // 16384

    hipLaunchKernelGGL(sparse_gate_topk, dim3(rows), dim3(NTHR), 0, stream,
                       x, y, mask);
}